// HSGeneratorLoss_76158360093210
// MI455X (gfx1250) — compile-verified
//
#include <hip/hip_runtime.h>
#include <math.h>

// MI455X / gfx1250 implementation of HSGeneratorLoss.
// Pairwise sq-distance tiles computed on the matrix unit:
//   V_WMMA_F32_16X16X4_F32 with A=[x_j,y_j,|p_j|^2,1], B=[-2x_i,-2y_i,1,|p_i|^2]
// -> D[m][n] = squared distance, full f32 precision.
// Top-k is done on raw sq (monotonic), sqrt only on the 4 survivors.

typedef float v2f __attribute__((ext_vector_type(2)));
typedef float v8f __attribute__((ext_vector_type(8)));

#define EPSV 1e-6f
#define N_PTS 2048
#define B_SZ 4
#define SORT_N 8192
#define HUGEF 3.4e38f

__device__ __forceinline__ float fsqrt_(float x) { return __builtin_amdgcn_sqrtf(x); }
__device__ __forceinline__ float frcp_(float x)  { return __builtin_amdgcn_rcpf(x); }
__device__ __forceinline__ float frsq_(float x)  { return __builtin_amdgcn_rsqf(x); }
__device__ __forceinline__ float fexp2_(float x) { return __builtin_amdgcn_exp2f(x); }

// ---------------------------------------------------------------------------
// Pass 1: WMMA pairwise sq distances -> kNN(3 incl self), 4th NN (excl self),
// feasibility overlap sum (DOFEAS only). One wave owns a 16-query column tile
// and sweeps all 128 j row tiles. C layout: lane l holds D[v+8*(l>>4)][l&15].
// ---------------------------------------------------------------------------
template <bool DOFEAS>
__global__ __launch_bounds__(256) void pass1_knn_wmma(
    const float* __restrict__ img, float* __restrict__ knnOut,
    float* __restrict__ kthOut, float* __restrict__ accOverlap)
{
  __shared__ __align__(16) float xs[N_PTS];
  __shared__ __align__(16) float ys[N_PTS];
  __shared__ __align__(16) float pp[N_PTS];
  __shared__ __align__(16) float ar[N_PTS];
  const int b = blockIdx.y;
  const float* p = img + (size_t)b * N_PTS * 3;
  __builtin_prefetch(p + threadIdx.x * 8, 0, 0);
  for (int n = threadIdx.x; n < N_PTS; n += 256) {
    float x = p[n * 3 + 0], y = p[n * 3 + 1], r = p[n * 3 + 2];
    xs[n] = x; ys[n] = y; pp[n] = x * x + y * y; ar[n] = fabsf(r);
  }
  __syncthreads();

  const int lane = threadIdx.x & 31;
  const int wave = threadIdx.x >> 5;
  const int i0   = (blockIdx.x * 8 + wave) * 16;
  const int nIdx = lane & 15;
  const int hi   = lane >> 4;          // lane half: selects K pair / M+8
  const int jhi  = hi * 8;
  const int iq   = i0 + nIdx;          // query point for this lane
  const float xi = xs[iq], yi = ys[iq], ppi = pp[iq], ari = ar[iq];

  // B fragment (4x16): lane l, vgpr v holds B[k = 2*hi + v][n = l&15]
  v2f bf;
  bf[0] = hi ? 1.0f : (-2.0f * xi);
  bf[1] = hi ? ppi  : (-2.0f * yi);

  // A-fragment source pointer chosen ONCE per lane -> unconditional ds loads
  const float* a0 = hi ? pp : xs;

  float t0 = HUGEF, t1 = HUGEF, t2 = HUGEF, t3 = HUGEF; // top-4 smallest sq
  float ov = 0.0f;

  const int NT = N_PTS / 16;
  v2f af; float arr8[8];
  {  // prologue: tile 0 operands
    const int jm = nIdx;
    float yv = ys[jm];
    af[0] = a0[jm];
    af[1] = hi ? 1.0f : yv;
    if (DOFEAS) {
#pragma unroll
      for (int v = 0; v < 8; ++v) arr8[v] = ar[jhi + v];
    }
  }

  for (int jt = 0; jt < NT; ++jt) {
    const int j0 = jt * 16;
    // software pipeline: issue next tile's LDS loads before consuming WMMA
    const int jtn = (jt + 1 < NT) ? (jt + 1) : jt;
    v2f afn; float arrn[8];
    {
      const int jm = jtn * 16 + nIdx;
      float yv = ys[jm];
      afn[0] = a0[jm];
      afn[1] = hi ? 1.0f : yv;
      if (DOFEAS) {
#pragma unroll
        for (int v = 0; v < 8; ++v) arrn[v] = ar[jtn * 16 + jhi + v];
      }
    }

    v8f c = {0.f, 0.f, 0.f, 0.f, 0.f, 0.f, 0.f, 0.f};
    c = __builtin_amdgcn_wmma_f32_16x16x4_f32(false, af, false, bf,
                                              (short)0, c, false, false);
#pragma unroll
    for (int v = 0; v < 8; ++v) {
      const int j = j0 + jhi + v;
      float sq = c[v];
      float nd = (j == iq) ? HUGEF : sq;   // mask self (branchless)
      float lo;
      lo = fminf(t0, nd); nd = fmaxf(t0, nd); t0 = lo;
      lo = fminf(t1, nd); nd = fmaxf(t1, nd); t1 = lo;
      lo = fminf(t2, nd); nd = fmaxf(t2, nd); t2 = lo;
      lo = fminf(t3, nd);                     t3 = lo;
      if (DOFEAS) {
        float d0 = fsqrt_(fmaxf(sq, 0.0f));
        float contrib = fmaxf(ari + arr8[v] - d0 - 1e-4f, 0.0f);
        ov += ((j < iq) && (sq > 0.0f)) ? contrib : 0.0f;
      }
    }
    af = afn;
    if (DOFEAS) {
#pragma unroll
      for (int v = 0; v < 8; ++v) arr8[v] = arrn[v];
    }
  }

  // merge the two lane halves (lane ^ 16 covered complementary j residues)
  {
    float o0 = __shfl_xor(t0, 16);
    float o1 = __shfl_xor(t1, 16);
    float o2 = __shfl_xor(t2, 16);
    float o3 = __shfl_xor(t3, 16);
    float ins[4] = {o0, o1, o2, o3};
#pragma unroll
    for (int t = 0; t < 4; ++t) {
      float nd = ins[t], lo;
      lo = fminf(t0, nd); nd = fmaxf(t0, nd); t0 = lo;
      lo = fminf(t1, nd); nd = fmaxf(t1, nd); t1 = lo;
      lo = fminf(t2, nd); nd = fmaxf(t2, nd); t2 = lo;
      lo = fminf(t3, nd);                     t3 = lo;
    }
  }

  if (DOFEAS) {
    ov += __shfl_xor(ov, 1);
    ov += __shfl_xor(ov, 2);
    ov += __shfl_xor(ov, 4);
    ov += __shfl_xor(ov, 8);
    ov += __shfl_xor(ov, 16);
    if (lane == 0) atomicAdd(accOverlap, ov);
  }

  if (hi == 0) {
    // convert surviving sq -> distances
    float d0 = fsqrt_(t0 + EPSV);
    float d1 = fsqrt_(t1 + EPSV);
    float d2 = fsqrt_(t2 + EPSV);
    float d3 = fsqrt_(t3 + EPSV);
    // kNN-3 including self distance (exactly sqrt(EPS) in the reference)
    float s = sqrtf(EPSV);
    float nd = s, lo, k0, k1, k2;
    lo = fminf(d0, nd); nd = fmaxf(d0, nd); k0 = lo;
    lo = fminf(d1, nd); nd = fmaxf(d1, nd); k1 = lo;
    lo = fminf(d2, nd);                     k2 = lo;
    float* o = knnOut + ((size_t)b * N_PTS + iq) * 3;
    o[0] = k0; o[1] = k1; o[2] = k2;
    if (kthOut) kthOut[b * N_PTS + iq] = d3;
  }
}

// ---------------------------------------------------------------------------
// Pass 2: hexatic order (k=4). e^{i4theta} via two complex squarings,
// sigmoid via v_exp (exp2), fully branchless inner loop.
// ---------------------------------------------------------------------------
__global__ __launch_bounds__(256) void pass2_hexatic(
    const float* __restrict__ img, const float* __restrict__ kthArr,
    float* __restrict__ accPsi)
{
  __shared__ __align__(16) float xs[N_PTS];
  __shared__ __align__(16) float ys[N_PTS];
  __shared__ float red[256];
  const int b = blockIdx.y;
  const float* p = img + (size_t)b * N_PTS * 3;
  __builtin_prefetch(p + threadIdx.x * 8, 0, 0);
  for (int n = threadIdx.x; n < N_PTS; n += 256) {
    xs[n] = p[n * 3 + 0]; ys[n] = p[n * 3 + 1];
  }
  __syncthreads();

  const int i = blockIdx.x * 256 + threadIdx.x;
  const float xi = xs[i], yi = ys[i];
  const float kth = kthArr[b * N_PTS + i];
  const float sigma = fmaxf(0.1f * fmaxf(kth, EPSV), EPSV);
  const float invsig = frcp_(sigma);
  const float L2E = 1.44269504088896340736f;

  float re = 0.f, im = 0.f, wsum = 0.f;
#pragma unroll 4
  for (int j = 0; j < N_PTS; ++j) {
    float dx = xi - xs[j];
    float dy = yi - ys[j];
    float dxa = dx + ((fabsf(dx) < EPSV) ? EPSV : 0.0f);
    float d = fsqrt_(dx * dx + dy * dy + EPSV);
    float arg = (kth - d) * invsig;
    arg = fminf(fmaxf(arg, -50.f), 50.f);
    float w = frcp_(1.0f + fexp2_(-arg * L2E));
    w = (j == i) ? 0.0f : w;          // eye -> w ~ sigmoid(-50) ~ 0
    float r2 = dxa * dxa + dy * dy;
    float inv = frsq_(r2);
    float zr = dxa * inv, zi = dy * inv;
    float z2r = zr * zr - zi * zi, z2i = 2.f * zr * zi;
    float z4r = z2r * z2r - z2i * z2i, z4i = 2.f * z2r * z2i;
    re = fmaf(w, z4r, re); im = fmaf(w, z4i, im); wsum += w;
  }
  float mag = fsqrt_(re * re + im * im) * frcp_(fmaxf(wsum, EPSV));

  red[threadIdx.x] = mag;
  __syncthreads();
  for (int s = 128; s > 0; s >>= 1) {
    if (threadIdx.x < s) red[threadIdx.x] += red[threadIdx.x + s];
    __syncthreads();
  }
  if (threadIdx.x == 0) atomicAdd(accPsi, red[0]);
}

// ---------------------------------------------------------------------------
// Feasibility denominator: sum |r| over all fake points.
// ---------------------------------------------------------------------------
__global__ __launch_bounds__(256) void reduce_absr(
    const float* __restrict__ fake, float* __restrict__ out)
{
  __shared__ float red[256];
  float s = 0.f;
  for (int n = threadIdx.x; n < B_SZ * N_PTS; n += 256)
    s += fabsf(fake[n * 3 + 2]);
  red[threadIdx.x] = s;
  __syncthreads();
  for (int t = 128; t > 0; t >>= 1) {
    if (threadIdx.x < t) red[threadIdx.x] += red[threadIdx.x + t];
    __syncthreads();
  }
  if (threadIdx.x == 0) *out = red[0];
}

// ---------------------------------------------------------------------------
// Bitonic sort (8192-wide, +INF pad) of the 14 arrays jnp.quantile needs.
// blockIdx: 0 fake_z, 1 real_z, 2 fake_x, 3 fake_y, 4 real_x, 5 real_y,
//           6..9 knn_real[b], 10..13 knn_fake[b].
// ---------------------------------------------------------------------------
__global__ __launch_bounds__(1024) void sort_quantile_arrays(
    const float* __restrict__ fake, const float* __restrict__ realp,
    const float* __restrict__ knnF, const float* __restrict__ knnR,
    float* __restrict__ sortedOut)
{
  __shared__ float s[SORT_N];
  const int a = blockIdx.x;
  const float* src; int stride, off, cnt;
  if (a < 6) {
    off = (a < 2) ? 2 : ((a == 2 || a == 4) ? 0 : 1);
    src = (a == 0 || a == 2 || a == 3) ? fake : realp;
    stride = 3; cnt = B_SZ * N_PTS;
  } else if (a < 10) {
    src = knnR + (size_t)(a - 6) * (N_PTS * 3); stride = 1; off = 0; cnt = N_PTS * 3;
  } else {
    src = knnF + (size_t)(a - 10) * (N_PTS * 3); stride = 1; off = 0; cnt = N_PTS * 3;
  }
  for (int idx = threadIdx.x; idx < SORT_N; idx += 1024)
    s[idx] = (idx < cnt) ? src[(size_t)idx * stride + off] : 3.0e38f;
  __syncthreads();

  for (int k = 2; k <= SORT_N; k <<= 1) {
    for (int j = k >> 1; j > 0; j >>= 1) {
      for (int idx = threadIdx.x; idx < SORT_N; idx += 1024) {
        int ixj = idx ^ j;
        if (ixj > idx) {
          float x = s[idx], y = s[ixj];
          bool up = (idx & k) == 0;
          if (up ? (x > y) : (x < y)) { s[idx] = y; s[ixj] = x; }
        }
      }
      __syncthreads();
    }
  }
  float* o = sortedOut + (size_t)a * SORT_N;
  for (int idx = threadIdx.x; idx < cnt; idx += 1024) o[idx] = s[idx];
}

// ---------------------------------------------------------------------------
// Final scalar combine (single thread; trivial amount of work).
// ---------------------------------------------------------------------------
__device__ __forceinline__ float quantile_sorted(const float* s, int n, float q) {
  float pos = q * (float)(n - 1);
  int lo = (int)floorf(pos);
  if (lo >= n - 1) return s[n - 1];
  float f = pos - (float)lo;
  return s[lo] + (s[lo + 1] - s[lo]) * f;
}

__global__ void init_accum(float* acc) {
  if (threadIdx.x < 4) acc[threadIdx.x] = 0.f;
}

__global__ void final_combine(const float* __restrict__ sorted,
                              const float* __restrict__ acc,
                              const float* __restrict__ fake_outputs,
                              float* __restrict__ out)
{
  if (threadIdx.x != 0) return;
  const float qs7[7] = {0.05f, 0.1f, 0.25f, 0.5f, 0.75f, 0.9f, 0.95f};
  const float qs5[5] = {0.05f, 0.25f, 0.5f, 0.75f, 0.95f};
  const float qs3[3] = {0.05f, 0.5f, 0.95f};
  const int n8 = SORT_N, n6 = N_PTS * 3;
  const float* fz = sorted + 0 * SORT_N;
  const float* rz = sorted + 1 * SORT_N;
  const float* fx = sorted + 2 * SORT_N;
  const float* fy = sorted + 3 * SORT_N;
  const float* rx = sorted + 4 * SORT_N;
  const float* ry = sorted + 5 * SORT_N;

  float loss = 0.f;

  // radius loss
  float rl = 0.f;
  for (int q = 0; q < 7; ++q) {
    float d = quantile_sorted(fz, n8, qs7[q]) - quantile_sorted(rz, n8, qs7[q]);
    rl += d * d;
  }
  loss += rl / 7.f;

  // physical feasibility: overlap_sum / (N * sum|r|)
  loss += acc[0] / ((float)N_PTS * acc[2]);

  // gan loss
  float g = 0.f;
  for (int b = 0; b < B_SZ; ++b) {
    float pv = fake_outputs[b];
    float lp = fmaxf(logf(pv), -100.f);
    float l1 = fmaxf(logf(1.f - pv), -100.f);
    g += 0.9f * lp + 0.1f * l1;
  }
  loss += -(g / (float)B_SZ);

  // grid density
  float lx = 0.f, ly = 0.f;
  for (int q = 0; q < 5; ++q) {
    float dxq = quantile_sorted(fx, n8, qs5[q]) - quantile_sorted(rx, n8, qs5[q]);
    float dyq = quantile_sorted(fy, n8, qs5[q]) - quantile_sorted(ry, n8, qs5[q]);
    lx += dxq * dxq; ly += dyq * dyq;
  }
  loss += (lx / 5.f + ly / 5.f) * 0.5f;

  // distance loss (per-batch quantiles over N*3 kNN dists)
  float dl = 0.f;
  for (int q = 0; q < 3; ++q)
    for (int b = 0; b < B_SZ; ++b) {
      float rq = quantile_sorted(sorted + (size_t)(6 + b) * SORT_N, n6, qs3[q]);
      float fq = quantile_sorted(sorted + (size_t)(10 + b) * SORT_N, n6, qs3[q]);
      float d = fq - rq;
      dl += d * d;
    }
  loss += dl / 12.f;

  // grid order
  loss += -(acc[1] / (float)(B_SZ * N_PTS));

  out[0] = loss;
}

// ---------------------------------------------------------------------------
extern "C" void kernel_launch(void* const* d_in, const int* in_sizes, int n_in,
                              void* d_out, int out_size, void* d_ws, size_t ws_size,
                              hipStream_t stream)
{
  (void)in_sizes; (void)n_in; (void)out_size; (void)ws_size;
  const float* realp = (const float*)d_in[0];   // real_images (4,2048,3)
  const float* fake  = (const float*)d_in[1];   // fake_images (4,2048,3)
  const float* fouts = (const float*)d_in[2];   // fake_outputs (4,1)
  float* outp = (float*)d_out;

  float* W      = (float*)d_ws;
  float* knnF   = W;                          // 4*2048*3 = 24576
  float* knnR   = W + 24576;                  // 24576
  float* kth    = W + 49152;                  // 8192
  float* sorted = W + 57344;                  // 14 * 8192 = 114688
  float* acc    = W + 57344 + 14 * SORT_N;    // [0]=overlap [1]=psi [2]=abs_r

  init_accum<<<1, 32, 0, stream>>>(acc);
  pass1_knn_wmma<true><<<dim3(16, B_SZ), 256, 0, stream>>>(fake, knnF, kth, acc + 0);
  pass1_knn_wmma<false><<<dim3(16, B_SZ), 256, 0, stream>>>(realp, knnR, nullptr, acc + 0);
  pass2_hexatic<<<dim3(8, B_SZ), 256, 0, stream>>>(fake, kth, acc + 1);
  reduce_absr<<<1, 256, 0, stream>>>(fake, acc + 2);
  sort_quantile_arrays<<<14, 1024, 0, stream>>>(fake, realp, knnF, knnR, sorted);
  final_combine<<<1, 32, 0, stream>>>(sorted, acc, fouts, outp);
}